// ST2FLS_TSK_38826504356510
// MI455X (gfx1250) — compile-verified
//
#include <hip/hip_runtime.h>
#include <math.h>

// Problem constants (from reference)
#define RN 32          // rules
#define AN 8           // antecedents
#define KF 28          // feature rows: [x(8), x^2(8), |x|(8), 1, pad3]
#define COLS 128       // 4 quantities (EL, EU, lin, amp) x 32 rules
#define TILE_M 16      // samples per wave tile (WMMA M)
#define WAVES 4        // waves per block (wave32)

typedef __attribute__((ext_vector_type(2))) float v2f;
typedef __attribute__((ext_vector_type(8))) float v8f;

// Pair-interleaved weight layout: rows (2p, 2p+1) of column `col` are the two
// adjacent floats at Wb[(p*COLS + col)*2 + {0,1}]  ->  one ds_load_b64 gives a
// WMMA B operand (K = 4c+2h, 4c+2h+1) directly in an aligned VGPR pair.
__host__ __device__ __forceinline__ int wpos(int r, int col) {
  return ((r >> 1) * COLS + col) * 2 + (r & 1);
}

// ---------------------------------------------------------------------------
// Setup: fold FRB_weights/C/S into Wb (pair-interleaved, KF x COLS floats):
//   cols  0..31 : EL coeffs (lower-firing exponent, w = 0.5*max(1/s1^2,1/s2^2))
//   cols 32..63 : EU coeffs (upper-firing exponent, w = 0.5*min(...))
//   cols 64..95 : lin  (C)        cols 96..127: amp (S)
// Reproduces reference's overlapping slices: m=W[8j+k], s1=W[8j+k+1], s2=W[8j+k+2]
// ---------------------------------------------------------------------------
__global__ __launch_bounds__(128) void st2fls_setup(
    const float* __restrict__ W, const float* __restrict__ C,
    const float* __restrict__ S, float* __restrict__ Wb) {
  const int col = threadIdx.x;
  if (col >= COLS) return;
  #pragma unroll
  for (int k = 0; k < KF; ++k) Wb[wpos(k, col)] = 0.0f;

  if (col < 64) {                       // EL (col<32) or EU
    const bool isEL = col < 32;
    const int j = col & 31;
    float cst = 0.0f;
    #pragma unroll
    for (int k = 0; k < AN; ++k) {
      const float m  = W[AN * j + k];
      const float s1 = W[AN * j + k + 1];
      const float s2 = W[AN * j + k + 2];
      const float i1 = 1.0f / (s1 * s1);
      const float i2 = 1.0f / (s2 * s2);
      const float w  = 0.5f * (isEL ? fmaxf(i1, i2) : fminf(i1, i2));
      Wb[wpos(8 + k, col)] = w;              // x^2 coefficient
      Wb[wpos(k, col)]     = -2.0f * m * w;  // x coefficient
      cst += m * m * w;                      // constant
    }
    Wb[wpos(24, col)] = cst;
  } else if (col < 96) {                // lin: x @ C[:,1:].T + C[:,0]
    const int j = col - 64;
    #pragma unroll
    for (int k = 0; k < AN; ++k) Wb[wpos(k, col)] = C[j * 9 + 1 + k];
    Wb[wpos(24, col)] = C[j * 9];
  } else {                              // amp: |x| @ S[:,1:].T + S[:,0]
    const int j = col - 96;
    #pragma unroll
    for (int k = 0; k < AN; ++k) Wb[wpos(16 + k, col)] = S[j * 9 + 1 + k];
    Wb[wpos(24, col)] = S[j * 9];
  }
}

// ------------------------- wave32 cross-lane helpers -----------------------
__device__ __forceinline__ float wsum(float v) {
  #pragma unroll
  for (int o = 16; o > 0; o >>= 1) v += __shfl_xor(v, o, 32);
  return v;
}
__device__ __forceinline__ float wmin(float v) {
  #pragma unroll
  for (int o = 16; o > 0; o >>= 1) v = fminf(v, __shfl_xor(v, o, 32));
  return v;
}
__device__ __forceinline__ float wmax(float v) {
  #pragma unroll
  for (int o = 16; o > 0; o >>= 1) v = fmaxf(v, __shfl_xor(v, o, 32));
  return v;
}
__device__ __forceinline__ float wscan(float v, int lane) {  // inclusive
  #pragma unroll
  for (int o = 1; o < 32; o <<= 1) {
    float t = __shfl_up(v, o, 32);
    if (lane >= o) v += t;
  }
  return v;
}
// Ascending bitonic sort of 32 (key,payload) pairs, one per lane.
__device__ __forceinline__ void bsort(float& key, float& pay, int lane) {
  #pragma unroll
  for (int k = 2; k <= 32; k <<= 1) {
    #pragma unroll
    for (int j = k >> 1; j > 0; j >>= 1) {
      const float ok = __shfl_xor(key, j, 32);
      const float op = __shfl_xor(pay, j, 32);
      const bool up      = (lane & k) == 0;
      const bool lowlane = (lane & j) == 0;
      const bool keepMin = (up == lowlane);
      const bool take = keepMin ? (ok < key) : (ok > key);
      if (take) { key = ok; pay = op; }
    }
  }
}

// ---------------------------------------------------------------------------
// Main kernel: WMMA front-end (features x weights) + wave-parallel KM.
// ---------------------------------------------------------------------------
__global__ __launch_bounds__(32 * WAVES) void st2fls_main(
    const float* __restrict__ X, const float* __restrict__ Wb,
    float* __restrict__ out, int n) {
  __shared__ float Bsh[KF * COLS];                 // 14 KB, shared weights
  __shared__ float stage[WAVES][TILE_M * COLS];    // 32 KB, per-wave D tiles

  const int tid = threadIdx.x;
  {  // vectorized one-time fill (both sides 16B-aligned, KF*COLS % 4 == 0)
    const float4* src = reinterpret_cast<const float4*>(Wb);
    float4* dst = reinterpret_cast<float4*>(Bsh);
    for (int i = tid; i < (KF * COLS) / 4; i += 32 * WAVES) dst[i] = src[i];
  }
  __syncthreads();

  const int lane = tid & 31;
  const int w    = tid >> 5;
  const bool hi  = lane >= 16;
  const int base = (blockIdx.x * WAVES + w) * TILE_M;

  // ---- load this lane's sample row (lanes 0-15 / 16-31 mirror samples) ----
  int row = base + (lane & 15);
  if (row >= n) row = n - 1;               // clamp; keep EXEC full for WMMA
  const float4* xr = reinterpret_cast<const float4*>(X + (size_t)row * AN);
  const float4 x0 = xr[0], x1 = xr[1];
  float fx[8] = {x0.x, x0.y, x0.z, x0.w, x1.x, x1.y, x1.z, x1.w};
  float fq[8], fa[8];
  #pragma unroll
  for (int k = 0; k < 8; ++k) { fq[k] = fx[k] * fx[k]; fa[k] = fabsf(fx[k]); }

  // feature value for compile-time K (folded after unroll)
  auto feat = [&](int K) -> float {
    if (K < 8)  return fx[K];
    if (K < 16) return fq[K - 8];
    if (K < 24) return fa[K - 16];
    return (K == 24) ? 1.0f : 0.0f;
  };

  // ---- A operands: 7 K-chunks, ISA 16x4 f32 layout (lanes16-31 carry K+2) --
  v2f Av[7];
  #pragma unroll
  for (int c = 0; c < 7; ++c) {
    const float l0 = feat(4 * c + 0), l1 = feat(4 * c + 1);
    const float h0 = feat(4 * c + 2), h1 = feat(4 * c + 3);
    Av[c].x = hi ? h0 : l0;
    Av[c].y = hi ? h1 : l1;
  }

  // which K-chunks contribute to each 16-col output tile (zero blocks skipped)
  // tiles 0-3: EL/EU need x, x^2, const; tiles 4-5: lin (x, const);
  // tiles 6-7: amp (|x|, const)
  constexpr bool used[8][7] = {
      {1,1,1,1,0,0,1}, {1,1,1,1,0,0,1}, {1,1,1,1,0,0,1}, {1,1,1,1,0,0,1},
      {1,1,0,0,0,0,1}, {1,1,0,0,0,0,1}, {0,0,0,0,1,1,1}, {0,0,0,0,1,1,1}};

  const int col  = lane & 15;
  const int hsel = hi ? 1 : 0;             // B 4x16 layout mirrors A's K split
  const v2f* Bp  = reinterpret_cast<const v2f*>(Bsh);

  #pragma unroll
  for (int t = 0; t < 8; ++t) {
    v8f acc = {};
    #pragma unroll
    for (int c = 0; c < 7; ++c) {
      if (!used[t][c]) continue;
      // rows (4c+2*hsel, 4c+2*hsel+1) of column col+16t: one b64 LDS load
      const v2f b = Bp[(2 * c + hsel) * COLS + col + 16 * t];
      acc = __builtin_amdgcn_wmma_f32_16x16x4_f32(
          false, Av[c], false, b, (short)0, acc, false, false);
    }
    // D layout: lane l VGPR v holds (M = v + 8*(l>=16), N = l&15)
    #pragma unroll
    for (int v = 0; v < 8; ++v)
      stage[w][(v + (hi ? 8 : 0)) * COLS + col + 16 * t] = acc[v];
  }

  // ---- Karnik-Mendel: lane j == rule j, one sample per iteration ----------
  float myOut = 0.0f;
  #pragma unroll 1
  for (int m = 0; m < TILE_M; ++m) {
    const float* sm = &stage[w][m * COLS];
    const float EL  = sm[lane];
    const float EU  = sm[32 + lane];
    const float lin = sm[64 + lane];
    const float amp = sm[96 + lane];
    const float Lf = __expf(-EL);          // lower firing = exp(-sum)
    const float Uf = __expf(-EU);          // upper firing
    const float c1 = lin - amp;
    const float c2 = lin + amp;

    // order-independent baselines
    const float s0l = wsum(c1 * Lf), t0l = wsum(Lf);
    const float s0r = wsum(c2 * Uf), t0r = wsum(Uf);

    // left endpoint: sort by c1 asc, carry d = U - L
    float kL = c1, dL = Uf - Lf;
    bsort(kL, dL, lane);
    const float cpL = wscan(kL * dL, lane);
    const float cdL = wscan(dL, lane);
    const float left = fminf(s0l / t0l, wmin((s0l + cpL) / (t0l + cdL)));

    // right endpoint: sort by c2 asc, carry d = L - U
    float kR = c2, dR = Lf - Uf;
    bsort(kR, dR, lane);
    const float cpR = wscan(kR * dR, lane);
    const float cdR = wscan(dR, lane);
    const float right = fmaxf(s0r / t0r, wmax((s0r + cpR) / (t0r + cdR)));

    const float res = 0.5f * (left + right);
    if (lane == m) myOut = res;            // lane m keeps sample m's result
  }

  if (lane < 16) {
    const int r = base + lane;
    if (r < n) out[r] = myOut;             // coalesced 64B store per wave
  }
}

// ---------------------------------------------------------------------------
extern "C" void kernel_launch(void* const* d_in, const int* in_sizes, int n_in,
                              void* d_out, int out_size, void* d_ws, size_t ws_size,
                              hipStream_t stream) {
  const float* X = (const float*)d_in[0];   // [N, 8]
  const float* W = (const float*)d_in[1];   // [768]
  const float* C = (const float*)d_in[2];   // [32, 9]
  const float* S = (const float*)d_in[3];   // [32, 9]
  float* out = (float*)d_out;
  float* Wb  = (float*)d_ws;                // 28*128 floats = 14336 B
  const int n = in_sizes[0] / AN;

  st2fls_setup<<<1, 128, 0, stream>>>(W, C, S, Wb);

  const int samplesPerBlock = TILE_M * WAVES;
  const int blocks = (n + samplesPerBlock - 1) / samplesPerBlock;
  st2fls_main<<<blocks, 32 * WAVES, 0, stream>>>(X, Wb, out, n);
}